// StructNDeconv2D_d_with_g_22660247454233
// MI455X (gfx1250) — compile-verified
//
#include <hip/hip_runtime.h>
#include <hip/hip_bf16.h>
#include <stdint.h>

#if __has_builtin(__builtin_amdgcn_global_load_async_to_lds_b128)
#define USE_ASYNC_LDS 1
#else
#define USE_ASYNC_LDS 0
#endif

namespace {

constexpr int Kk   = 3;
constexpr int Bc   = 2;
constexpr int Cc   = 8;
constexpr int Hc   = 96;
constexpr int Wc   = 320;
constexpr int HOUTc = 191;
constexpr int WOUTc = 639;
constexpr float EPSc = 1e-20f;

#if USE_ASYNC_LDS
typedef int v4i_t __attribute__((vector_size(16)));
typedef v4i_t __attribute__((address_space(1)))* g_v4i_p;   // global
typedef v4i_t __attribute__((address_space(3)))* l_v4i_p;   // LDS
#endif

__device__ __forceinline__ float softplusf(float x) {
    // jax.nn.softplus: max(x,0) + log1p(exp(-|x|)) (stable form)
    return fmaxf(x, 0.0f) + log1pf(expf(-fabsf(x)));
}

// One block per (b*C+c, oy). Thread m produces output pixels ox=2m and ox=2m+1.
// Parity math: oy even -> only kernel row a=1 (input row oy/2) is live;
// oy odd -> rows a=0 (input row oy/2+1, dy=+2) and a=2 (input row oy/2, dy=-2),
// both always in bounds. ox=2m -> b=1 (j=m); ox=2m+1 -> b=0 (j=m+1, dx=+2) and
// b=2 (j=m, dx=-2), in bounds iff m<319.
__global__ __launch_bounds__(Wc) void StructNDeconv2D_d_with_g_kernel(
    const float* __restrict__ d,   const float* __restrict__ cd,
    const float* __restrict__ gx,  const float* __restrict__ cgx,
    const float* __restrict__ gy,  const float* __restrict__ cgy,
    const float* __restrict__ sw_raw, const float* __restrict__ wg_raw,
    float* __restrict__ d_out, float* __restrict__ cd_out)
{
    __shared__ float s_sw[Kk * Kk];
    __shared__ float s_wg;
    // tile[tensor*2 + row][col]; tensor: 0=d 1=cd 2=gx 3=cgx 4=gy 5=cgy
    // row: 0 = (a=0 row | the single even-oy row), 1 = a=2 row (odd oy only)
    __shared__ float tile[12][Wc];   // 15360 B

    const int bc  = blockIdx.x;          // b*C + c
    const int oy  = blockIdx.y;
    const int c   = bc & (Cc - 1);
    const int tid = (int)threadIdx.x;    // 0..319

    if (tid < Kk * Kk) s_sw[tid] = softplusf(sw_raw[c * Kk * Kk + tid]);
    if (tid == Kk * Kk) s_wg = softplusf(wg_raw[c]);

    const bool odd = (oy & 1) != 0;
    const int  n    = oy >> 1;
    const int  rowA = odd ? (n + 1) : n;   // a=0 row (odd) / a=1 row (even)
    const int  rowB = n;                   // a=2 row (odd only)

    const int base = bc * (Hc * Wc);
    const float* srcs[6] = { d + base, cd + base, gx + base,
                             cgx + base, gy + base, cgy + base };

    // ---------------- stage input rows into LDS ----------------
    // 80 float4 chunks per (tensor,row) plane; threads [0,80) do row A,
    // threads [80,160) do row B (odd oy only).
    {
        const int r  = tid / 80;
        const int ch = tid % 80;
        const int nrows = odd ? 2 : 1;
        const int ntens = odd ? 6 : 4;           // gy/cgy unused when dy==0
        const int roff  = ((r == 0) ? rowA : rowB) * Wc + ch * 4;
        if (r < nrows) {
            #pragma unroll
            for (int t = 0; t < 6; ++t) {
                if (t < ntens) {
#if USE_ASYNC_LDS
                    // gfx1250 async copy: global -> LDS, tracked by ASYNCcnt
                    g_v4i_p gp = (g_v4i_p)(uintptr_t)(const void*)(srcs[t] + roff);
                    l_v4i_p lp = (l_v4i_p)&tile[t * 2 + r][ch * 4];
                    __builtin_amdgcn_global_load_async_to_lds_b128(gp, lp, 0, 0);
#else
                    const float4 v = *reinterpret_cast<const float4*>(srcs[t] + roff);
                    *reinterpret_cast<float4*>(&tile[t * 2 + r][ch * 4]) = v;
#endif
                }
            }
        }
    }
#if USE_ASYNC_LDS
#if __has_builtin(__builtin_amdgcn_s_wait_asynccnt)
    __builtin_amdgcn_s_wait_asynccnt(0);
#else
    asm volatile("s_wait_asynccnt 0x0" ::: "memory");
#endif
#endif
    __syncthreads();   // gfx1250 split barrier

    const float wg      = s_wg;
    const float one_pwg = 1.0f + wg;
    float swl[Kk * Kk];
    #pragma unroll
    for (int p = 0; p < Kk * Kk; ++p) swl[p] = s_sw[p];

    const int m     = tid;
    const int obase = (bc * HOUTc + oy) * WOUTc;

    float outd0 = 0.0f, outc0 = 0.0f, outd1 = 0.0f, outc1 = 0.0f;

    if (!odd) {
        // planes: d=0, cd=2, gx=4, cgx=6 (row A only)
        const float d_m  = tile[0][m];
        const float cd_m = tile[2][m];
        {   // ox = 2m : center tap p=4 (dx=dy=0)
            const float cp  = cd_m / one_pwg;          // cgprop==0, gprop==0
            const float den = swl[4] * cp;
            outd0 = (swl[4] * (cp * d_m)) / (den + EPSc);
            outc0 = den / (swl[4] + EPSc);
        }
        if (m < Wc - 1) {   // ox = 2m+1 : taps p=3 (j=m+1,dx=+2), p=5 (j=m,dx=-2)
            const float d_m1   = tile[0][m + 1];
            const float cd_m1  = tile[2][m + 1];
            const float gx_m1  = tile[4][m + 1];
            const float cgx_m1 = tile[6][m + 1];
            const float gx_m   = tile[4][m];
            const float cgx_m  = tile[6][m];

            const float dp3 = d_m1 * (1.0f + 2.0f * gx_m1);
            const float cp3 = cd_m1 * (1.0f + wg * ((2.0f * cgx_m1) / (2.0f + EPSc))) / one_pwg;
            const float dp5 = d_m * (1.0f - 2.0f * gx_m);
            const float cp5 = cd_m * (1.0f + wg * ((2.0f * cgx_m) / (2.0f + EPSc))) / one_pwg;

            const float nom  = swl[3] * (cp3 * dp3) + swl[5] * (cp5 * dp5);
            const float den  = swl[3] * cp3 + swl[5] * cp5;
            const float cden = swl[3] + swl[5];
            outd1 = nom / (den + EPSc);
            outc1 = den / (cden + EPSc);
        }
    } else {
        // row A = a=0 (dy=+2) planes t*2; row B = a=2 (dy=-2) planes t*2+1
        const float dA_m   = tile[0][m],  dB_m   = tile[1][m];
        const float cdA_m  = tile[2][m],  cdB_m  = tile[3][m];
        const float gyA_m  = tile[8][m],  gyB_m  = tile[9][m];
        const float cgyA_m = tile[10][m], cgyB_m = tile[11][m];
        {   // ox = 2m : taps p=1 (rowA), p=7 (rowB), j=m, dx=0
            const float dp1 = dA_m * (1.0f + 2.0f * gyA_m);
            const float cp1 = cdA_m * (1.0f + wg * ((2.0f * cgyA_m) / (2.0f + EPSc))) / one_pwg;
            const float dp7 = dB_m * (1.0f - 2.0f * gyB_m);
            const float cp7 = cdB_m * (1.0f + wg * ((2.0f * cgyB_m) / (2.0f + EPSc))) / one_pwg;

            const float nom  = swl[1] * (cp1 * dp1) + swl[7] * (cp7 * dp7);
            const float den  = swl[1] * cp1 + swl[7] * cp7;
            const float cden = swl[1] + swl[7];
            outd0 = nom / (den + EPSc);
            outc0 = den / (cden + EPSc);
        }
        if (m < Wc - 1) {   // ox = 2m+1 : taps p=0,2 (rowA), p=6,8 (rowB)
            const float dA_m1   = tile[0][m + 1],  dB_m1   = tile[1][m + 1];
            const float cdA_m1  = tile[2][m + 1],  cdB_m1  = tile[3][m + 1];
            const float gxA_m   = tile[4][m],      gxB_m   = tile[5][m];
            const float gxA_m1  = tile[4][m + 1],  gxB_m1  = tile[5][m + 1];
            const float cgxA_m  = tile[6][m],      cgxB_m  = tile[7][m];
            const float cgxA_m1 = tile[6][m + 1],  cgxB_m1 = tile[7][m + 1];
            const float gyA_m1  = tile[8][m + 1],  gyB_m1  = tile[9][m + 1];
            const float cgyA_m1 = tile[10][m + 1], cgyB_m1 = tile[11][m + 1];

            // p=0: dx=+2, dy=+2, j=m+1, row A
            const float dp0 = dA_m1 * (1.0f + (2.0f * gxA_m1 + 2.0f * gyA_m1));
            const float cp0 = cdA_m1 * (1.0f + wg * ((2.0f * cgxA_m1 + 2.0f * cgyA_m1) / (4.0f + EPSc))) / one_pwg;
            // p=2: dx=-2, dy=+2, j=m, row A
            const float dp2 = dA_m * (1.0f + (-2.0f * gxA_m + 2.0f * gyA_m));
            const float cp2 = cdA_m * (1.0f + wg * ((2.0f * cgxA_m + 2.0f * cgyA_m) / (4.0f + EPSc))) / one_pwg;
            // p=6: dx=+2, dy=-2, j=m+1, row B
            const float dp6 = dB_m1 * (1.0f + (2.0f * gxB_m1 - 2.0f * gyB_m1));
            const float cp6 = cdB_m1 * (1.0f + wg * ((2.0f * cgxB_m1 + 2.0f * cgyB_m1) / (4.0f + EPSc))) / one_pwg;
            // p=8: dx=-2, dy=-2, j=m, row B
            const float dp8 = dB_m * (1.0f + (-2.0f * gxB_m - 2.0f * gyB_m));
            const float cp8 = cdB_m * (1.0f + wg * ((2.0f * cgxB_m + 2.0f * cgyB_m) / (4.0f + EPSc))) / one_pwg;

            const float nom  = ((swl[0] * (cp0 * dp0) + swl[2] * (cp2 * dp2))
                                + swl[6] * (cp6 * dp6)) + swl[8] * (cp8 * dp8);
            const float den  = ((swl[0] * cp0 + swl[2] * cp2) + swl[6] * cp6) + swl[8] * cp8;
            const float cden = ((swl[0] + swl[2]) + swl[6]) + swl[8];
            outd1 = nom / (den + EPSc);
            outc1 = den / (cden + EPSc);
        }
    }

    const int o0 = obase + 2 * m;
    d_out[o0]  = outd0;
    cd_out[o0] = outc0;
    if (m < Wc - 1) {
        d_out[o0 + 1]  = outd1;
        cd_out[o0 + 1] = outc1;
    }
}

} // anonymous namespace

extern "C" void kernel_launch(void* const* d_in, const int* in_sizes, int n_in,
                              void* d_out, int out_size, void* d_ws, size_t ws_size,
                              hipStream_t stream) {
    (void)in_sizes; (void)n_in; (void)out_size; (void)d_ws; (void)ws_size;

    const float* d   = (const float*)d_in[0];
    const float* cd  = (const float*)d_in[1];
    const float* gx  = (const float*)d_in[2];
    const float* cgx = (const float*)d_in[3];
    const float* gy  = (const float*)d_in[4];
    const float* cgy = (const float*)d_in[5];
    const float* sw  = (const float*)d_in[6];
    const float* wgr = (const float*)d_in[7];

    float* out = (float*)d_out;
    const int N = Bc * Cc * HOUTc * WOUTc;

    dim3 grid(Bc * Cc, HOUTc);     // 16 x 191 blocks, 320 threads (10 wave32s)
    StructNDeconv2D_d_with_g_kernel<<<grid, Wc, 0, stream>>>(
        d, cd, gx, cgx, gy, cgy, sw, wgr, out, out + N);
}